// SamplingWrapper_11123965297296
// MI455X (gfx1250) — compile-verified
//
#include <hip/hip_runtime.h>
#include <stdint.h>
#include <limits.h>

// top-p sampling: ONE WAVE (32 lanes) per row of V=1024 logits.
// 8 waves / 256-thread block => 8 rows per block, zero block-level sync.
#define TEMP_INV 1.25f        // 1 / 0.8
#define TOP_P    0.9f
#define VOCAB    1024
#define WPB      8            // waves per block
#define NT       (WPB * 32)   // 256 threads
#define EPL      32           // elements per lane (VOCAB / 32)

typedef __attribute__((ext_vector_type(2))) float v2f;
typedef __attribute__((ext_vector_type(8))) float v8f;

__device__ __forceinline__ float bperm_f(int srcLane, float v) {
  int r = __builtin_amdgcn_ds_bpermute(srcLane << 2, __float_as_int(v));
  return __int_as_float(r);
}
__device__ __forceinline__ int bperm_i(int srcLane, int v) {
  return __builtin_amdgcn_ds_bpermute(srcLane << 2, v);
}

// Wave-wide sum of (x0 + x1) over all 32 lanes via the matrix pipe:
// A = 16x4 f32 (lane carries {x0, x1} => all 64 A slots used), B = ones(4x16)
// -> D[m,n] = rowsum(A,m). Per-lane sum of its 8 D rows covers M=0..7
// (lanes 0-15) or M=8..15 (lanes 16-31); xor-16 bpermute-add => total.
__device__ __forceinline__ float wave_sum_wmma(float x0, float x1, int lane) {
  v2f a; a[0] = x0;   a[1] = x1;
  v2f b; b[0] = 1.0f; b[1] = 1.0f;
  v8f c = {};
  v8f d = __builtin_amdgcn_wmma_f32_16x16x4_f32(
      /*neg_a=*/false, a, /*neg_b=*/false, b,
      /*c_mod=*/(short)0, c, /*reuse_a=*/false, /*reuse_b=*/false);
  float s = d[0] + d[1] + d[2] + d[3] + d[4] + d[5] + d[6] + d[7];
  s += bperm_f(lane ^ 16, s);
  return s;                                   // uniform across the wave
}

__device__ __forceinline__ float hash_u01(unsigned x) {
  x += 0x9E3779B9u;
  x ^= x >> 16; x *= 0x21F0AAADu;
  x ^= x >> 15; x *= 0x735A2D97u;
  x ^= x >> 15;
  return (float)(x >> 8) * (1.0f / 16777216.0f);
}

__global__ void __launch_bounds__(NT)
topp_sample_kernel(const int* __restrict__ codes,
                   const unsigned char* __restrict__ mask,
                   const float* __restrict__ logits,
                   int* __restrict__ out, int rows) {
  const int t    = (int)threadIdx.x;
  const int lane = t & 31;
  const int wave = t >> 5;
  const int row  = blockIdx.x * WPB + wave;
  if (row >= rows) return;                    // wave-uniform

  __shared__ __align__(16) float stage[WPB][VOCAB];

  // ---- CDNA5 async global->LDS staging (ASYNCcnt path) --------------------
  // INST_OFFSET is added to BOTH lds and global addresses, so one base pair
  // + 8 x b128 w/ offset:k*512 stages the whole 4KB row, 512B coalesced each.
  {
    unsigned     ldst = (unsigned)(uintptr_t)(&stage[wave][0]) +
                        (unsigned)(lane * 16);
    const char*  gsrc = (const char*)(logits + (size_t)row * VOCAB) + lane * 16;
    asm volatile(
        "global_load_async_to_lds_b128 %0, %1, off\n\t"
        "global_load_async_to_lds_b128 %0, %1, off offset:512\n\t"
        "global_load_async_to_lds_b128 %0, %1, off offset:1024\n\t"
        "global_load_async_to_lds_b128 %0, %1, off offset:1536\n\t"
        "global_load_async_to_lds_b128 %0, %1, off offset:2048\n\t"
        "global_load_async_to_lds_b128 %0, %1, off offset:2560\n\t"
        "global_load_async_to_lds_b128 %0, %1, off offset:3072\n\t"
        "global_load_async_to_lds_b128 %0, %1, off offset:3584\n\t"
        "s_wait_asynccnt 0"
        :
        : "v"(ldst), "v"(gsrc)
        : "memory");
  }
  // no barrier: each wave reads only the LDS region its own asyncs wrote

  // ---- LDS -> registers (8 x ds_load_b128), lane owns [lane*32, lane*32+32)
  float e[EPL];
  {
    const float4* src = (const float4*)(&stage[wave][lane * EPL]);
#pragma unroll
    for (int k = 0; k < EPL / 4; ++k) {
      float4 v4 = src[k];
      e[4 * k + 0] = v4.x; e[4 * k + 1] = v4.y;
      e[4 * k + 2] = v4.z; e[4 * k + 3] = v4.w;
    }
  }

  // ---- wave max (stable softmax) ------------------------------------------
  float mx = e[0];
#pragma unroll
  for (int j = 1; j < EPL; ++j) mx = fmaxf(mx, e[j]);
#pragma unroll
  for (int m = 16; m >= 1; m >>= 1) mx = fmaxf(mx, bperm_f(lane ^ m, mx));

  // ---- unnormalized probs q_i = exp((l_i - max)/T); q_max == 1.0 ----------
  float s0 = 0.0f, s1 = 0.0f;
#pragma unroll
  for (int j = 0; j < EPL; ++j) {
    e[j] = __expf((e[j] - mx) * TEMP_INV);
    if (j < EPL / 2) s0 += e[j]; else s1 += e[j];
  }
  const float Z      = wave_sum_wmma(s0, s1, lane);
  const float target = TOP_P * Z;

  // ---- binary-search top-p threshold theta in [0, 1] ----------------------
  // kept set {q >= theta}; want largest kept mass <= TOP_P * Z.
  float lo = 0.0f, hi = 1.000001f;
  for (int it = 0; it < 24; ++it) {
    float mid = 0.5f * (lo + hi);
    float p0 = 0.0f, p1 = 0.0f;
#pragma unroll
    for (int j = 0; j < EPL; ++j) {
      float k = (e[j] >= mid) ? e[j] : 0.0f;
      if (j < EPL / 2) p0 += k; else p1 += k;
    }
    float m = wave_sum_wmma(p0, p1, lane);    // uniform across wave
    if (m > target) lo = mid; else hi = mid;  // wave-uniform branch
  }
  float theta = hi;
  float p0 = 0.0f, p1 = 0.0f;
#pragma unroll
  for (int j = 0; j < EPL; ++j) {
    float k = (e[j] >= theta) ? e[j] : 0.0f;
    if (j < EPL / 2) p0 += k; else p1 += k;
  }
  float kmass = wave_sum_wmma(p0, p1, lane);
  if (!(kmass > 0.0f)) {        // even argmax alone exceeds TOP_P: keep it
    theta = 1.0f;               // q_argmax == exp(0) == 1.0 exactly
    p0 = p1 = 0.0f;
#pragma unroll
    for (int j = 0; j < EPL; ++j) {
      float k = (e[j] >= theta) ? e[j] : 0.0f;
      if (j < EPL / 2) p0 += k; else p1 += k;
    }
    kmass = wave_sum_wmma(p0, p1, lane);
  }

  // ---- categorical draw over kept set via wave prefix scan ----------------
  const float u = hash_u01((unsigned)row) * kmass;

  float ttot = 0.0f;
#pragma unroll
  for (int j = 0; j < EPL; ++j) ttot += (e[j] >= theta) ? e[j] : 0.0f;

  float sc = ttot;                            // wave inclusive scan
#pragma unroll
  for (int d = 1; d < 32; d <<= 1) {
    float n = bperm_f(lane - d, sc);
    if (lane >= d) sc += n;
  }
  const float before = sc - ttot;             // exclusive prefix for this lane

  int cand   = INT_MAX;                       // crossing index
  int firstk = INT_MAX;                       // first kept index (fallback)
#pragma unroll
  for (int j = 0; j < EPL; ++j) {
    if (e[j] >= theta && firstk == INT_MAX) firstk = lane * EPL + j;
  }
  if (before <= u && u < before + ttot) {
    float r = before;
#pragma unroll
    for (int j = 0; j < EPL; ++j) {
      if (e[j] >= theta) {
        r += e[j];
        if (r > u && cand == INT_MAX) cand = lane * EPL + j;
      }
    }
  }
#pragma unroll
  for (int m = 16; m >= 1; m >>= 1) {         // wave-min reductions
    cand   = min(cand,   bperm_i(lane ^ m, cand));
    firstk = min(firstk, bperm_i(lane ^ m, firstk));
  }

  if (lane == 0) {
    int sampled = (cand != INT_MAX) ? cand
                : (firstk != INT_MAX) ? firstk : 0;
    out[row] = mask[row] ? sampled : codes[row];
  }
}

extern "C" void kernel_launch(void* const* d_in, const int* in_sizes, int n_in,
                              void* d_out, int out_size, void* d_ws, size_t ws_size,
                              hipStream_t stream) {
  const int*           codes  = (const int*)d_in[0];
  const unsigned char* mask   = (const unsigned char*)d_in[1];
  const float*         logits = (const float*)d_in[2];
  int*                 out    = (int*)d_out;
  const int rows   = in_sizes[0];             // B*C*S = 57344, V = 1024
  const int blocks = (rows + WPB - 1) / WPB;
  topp_sample_kernel<<<blocks, NT, 0, stream>>>(codes, mask, logits, out, rows);
}